// PointPillarsScatter_34248069218718
// MI455X (gfx1250) — compile-verified
//
#include <hip/hip_runtime.h>

// PointPillars scatter for MI455X (gfx1250).
// Pure data movement: 256MB canvas zero-fill + 6.4M scattered 4B stores.
// Roofline @ 23.3 TB/s HBM: ~13 us total, dominated by the zero-fill.
// CDNA5 path: GLOBAL_STORE_ASYNC_FROM_LDS_B128 (ASYNCcnt DMA stores);
// non-temporal vmem for the scatter.

#define CANVAS_H 1024
#define CANVAS_W 1024
#define CANVAS_C 64

#define FILL_BLOCK   256                        // threads per block (8 waves)
#define FILL_UNROLL  8                          // async stores per thread
#define FILL_TILE    (FILL_BLOCK * FILL_UNROLL) // float4s per block = 2048 (32KB)

typedef float float4v __attribute__((ext_vector_type(4)));
typedef int   int4v   __attribute__((ext_vector_type(4)));

// Builtin parameter types (confirmed by round-1 diagnostic): async LDS<->global
// b128 builtins take pointers to 4 x i32 vectors in AS(1) (global) / AS(3) (LDS).
typedef __attribute__((address_space(1))) int4v gint4_t; // global int4
typedef __attribute__((address_space(3))) int4v lint4_t; // LDS int4

#if defined(__has_builtin)
#if __has_builtin(__builtin_amdgcn_global_store_async_from_lds_b128)
#define HAVE_ASYNC_LDS_STORE 1
#endif
#if __has_builtin(__builtin_amdgcn_s_wait_asynccnt)
#define HAVE_WAIT_ASYNCCNT 1
#endif
#endif

// ---------------------------------------------------------------------------
// Kernel 1: zero the 256MB canvas.
// Each block owns one contiguous 32KB tile. Full-tile blocks (the only kind
// for the 16M-float4 canvas) take a *block-uniform* scalar branch and issue
// FILL_UNROLL unconditional async DMA stores per thread, each sourced from the
// thread's private 16B chunk of a zeroed 4KB LDS tile. NOTE: the IOFFSET
// immediate applies to BOTH the LDS and global addresses on this instruction
// (ISA 08 §4.4), so global stepping must stay in the address VGPRs.
// ASYNCcnt hardware-stalls issue at 63 outstanding; wait once at the end.
// ---------------------------------------------------------------------------
__global__ void ppscatter_zero_kernel(float* __restrict__ out,
                                      unsigned n4) {
  int4v* o4 = (int4v*)out;
  const unsigned tile_base = blockIdx.x * FILL_TILE;
  const unsigned base = tile_base + threadIdx.x;

#if HAVE_ASYNC_LDS_STORE
  __shared__ int4v ztile[FILL_BLOCK]; // 4KB staging tile, 16B chunk per thread
  ztile[threadIdx.x] = (int4v)0;
  __syncthreads();

  // LDS source address as AS(3) pointer: generic->local is low-32 truncation.
  lint4_t* lsrc =
      (lint4_t*)(unsigned int)(unsigned long long)(&ztile[threadIdx.x]);
  gint4_t* gbase = (gint4_t*)(unsigned long long)(o4 + base);

  if (tile_base + FILL_TILE <= n4) {
    // Fast path: whole tile in range (uniform scalar branch, no EXEC churn).
#pragma unroll
    for (int k = 0; k < FILL_UNROLL; ++k)
      __builtin_amdgcn_global_store_async_from_lds_b128(
          gbase + k * FILL_BLOCK, lsrc, 0, 0);
  } else {
    // Tail tile: per-lane predicated stores (EXEC masking is legal for VMEM).
#pragma unroll
    for (int k = 0; k < FILL_UNROLL; ++k)
      if (base + (unsigned)k * FILL_BLOCK < n4)
        __builtin_amdgcn_global_store_async_from_lds_b128(
            gbase + k * FILL_BLOCK, lsrc, 0, 0);
  }

#if HAVE_WAIT_ASYNCCNT
  __builtin_amdgcn_s_wait_asynccnt(0);
#else
  asm volatile("s_wait_asynccnt 0" ::: "memory");
#endif
#else
  // Fallback: plain wide non-temporal stores, same tiling.
  const int4v z = (int4v)0;
#pragma unroll
  for (int k = 0; k < FILL_UNROLL; ++k) {
    const unsigned idx = base + (unsigned)k * FILL_BLOCK;
    if (idx < n4) __builtin_nontemporal_store(z, o4 + idx);
  }
#endif
}

// ---------------------------------------------------------------------------
// Kernel 2: scatter features into the canvas.
// One thread per (pillar, channel). feat[tid] is fully coalesced; the canvas
// store is random-address by construction, so use NT hints (canvas = 256MB >
// 192MB L2; no reuse). coords loads are wave-uniform (32 lanes share a pillar).
// ---------------------------------------------------------------------------
__global__ void ppscatter_scatter_kernel(const float* __restrict__ feat,
                                         const int* __restrict__ coords,
                                         float* __restrict__ out,
                                         int total) {
  int tid = blockIdx.x * blockDim.x + threadIdx.x;
  if (tid >= total) return;

  const int n = tid >> 6;   // pillar index
  const int c = tid & 63;   // channel index

  const int x = coords[2 * n + 0];
  const int y = coords[2 * n + 1];

  // canvas layout (C, H, W) with H = W = 1024: idx = c*2^20 + y*2^10 + x
  const unsigned idx =
      ((unsigned)c << 20) | ((unsigned)y << 10) | (unsigned)x;

  const float v = __builtin_nontemporal_load(feat + tid);
  __builtin_nontemporal_store(v, out + idx);
}

extern "C" void kernel_launch(void* const* d_in, const int* in_sizes, int n_in,
                              void* d_out, int out_size, void* d_ws, size_t ws_size,
                              hipStream_t stream) {
  const float* feat  = (const float*)d_in[0];   // (100000, 64) fp32
  const int* coords  = (const int*)d_in[1];     // (100000, 2) int
  float* out         = (float*)d_out;           // (1, 64, 1024, 1024) fp32

  // ---- zero fill: out_size floats -> out_size/4 float4s (64M -> 16M) -------
  const unsigned n4 = (unsigned)(out_size / 4);
  const unsigned fill_blocks = (n4 + FILL_TILE - 1) / FILL_TILE; // 8192
  ppscatter_zero_kernel<<<fill_blocks, FILL_BLOCK, 0, stream>>>(out, n4);

  // ---- scatter: one thread per (pillar, channel) ---------------------------
  const int total = in_sizes[0]; // N * 64 = 6,400,000
  const int blocks = (total + 255) / 256;
  ppscatter_scatter_kernel<<<blocks, 256, 0, stream>>>(feat, coords, out, total);
}